// MiniMaxM2MoeRoutingMethod_66340064854662
// MI455X (gfx1250) — compile-verified
//
#include <hip/hip_runtime.h>
#include <stdint.h>

#define EXPERTS 256
#define TOPK 8
#define TOK_PER_WAVE 4
#define WAVES_PER_BLOCK 8
#define NEG_INF (-3.402823466e38f)

// ---------------------------------------------------------------------------
// MoE sigmoid routing, biased top-8, normalized unbiased weights.
// Bandwidth floor: 131072*256*4B = 128MiB read @ 23.3 TB/s ~= 6us, so the
// kernel streams rows HBM->LDS with CDNA5 async DMA (global_load_async_to_lds
// + s_wait_asynccnt, double buffered) and keeps the per-token top-8 selection
// as lean as possible:
//   - lane l owns contiguous experts [8l, 8l+8): ballot-ctz lane order ==
//     expert index order, so tie-breaking is exactly jax.lax.top_k's
//     (value desc, index asc) with zero extra work.
//   - per round: 8-wide local argmax, 5-step value-only f32 max butterfly,
//     one ballot + ctz + readlane to resolve the winner, scalar-slot
//     invalidation. No (value,index,score) triple shuffles.
//   - sigmoid row is written back to the consumed LDS buffer; the 8 winner
//     weights are fetched at the end with a single divergent ds_load_b32.
// ---------------------------------------------------------------------------
__global__ __launch_bounds__(256) void moe_route_topk8_kernel(
    const float* __restrict__ logits,   // [T, 256]
    const float* __restrict__ bias,     // [256]
    int*   __restrict__ idx_out,        // [T, 8]
    float* __restrict__ w_out,          // [T, 8]
    int T)
{
    // 8 waves * 2 buffers * 256 floats = 16 KB
    __shared__ float lds[WAVES_PER_BLOCK * 2 * EXPERTS];

    const int lane      = threadIdx.x & 31;
    const int waveInBlk = threadIdx.x >> 5;
    const int wave      = blockIdx.x * WAVES_PER_BLOCK + waveInBlk;
    const int t0        = wave * TOK_PER_WAVE;

    float* wbuf = &lds[waveInBlk * (2 * EXPERTS)];
    const unsigned ldsByte0 = (unsigned)(uintptr_t)wbuf;       // buffer 0
    const unsigned ldsByte1 = ldsByte0 + EXPERTS * 4;          // buffer 1
    const uint64_t gbase    = (uint64_t)(uintptr_t)logits;

    // Selection bias for this lane's experts 8l..8l+7 (reused for 4 tokens).
    const float4 blo = ((const float4*)bias)[2 * lane];
    const float4 bhi = ((const float4*)bias)[2 * lane + 1];

    // One token row (1 KB) = two 512B async global->LDS transfers.
    auto issue_row = [&](unsigned ldsOff, int tok) {
        const unsigned g0 = (unsigned)tok * (EXPERTS * 4) + (unsigned)lane * 16;
        const unsigned g1 = g0 + 512;
        const unsigned l0 = ldsOff + (unsigned)lane * 16;
        const unsigned l1 = l0 + 512;
        asm volatile(
            "global_load_async_to_lds_b128 %0, %1, %4\n\t"
            "global_load_async_to_lds_b128 %2, %3, %4"
            :
            : "v"(l0), "v"(g0), "v"(l1), "v"(g1), "s"(gbase)
            : "memory");
    };

    if (t0 < T) issue_row(ldsByte0, t0);

    for (int i = 0; i < TOK_PER_WAVE; ++i) {
        const int cur = t0 + i;
        if (cur >= T) break;
        const bool have_next = (i + 1 < TOK_PER_WAVE) && (cur + 1 < T);
        if (have_next) issue_row((i & 1) ? ldsByte0 : ldsByte1, cur + 1);

        // Retire current row's 2 async ops; keep the next row in flight.
        if (have_next) asm volatile("s_wait_asynccnt 0x2" ::: "memory");
        else           asm volatile("s_wait_asynccnt 0x0" ::: "memory");

        float* buf = (i & 1) ? (wbuf + EXPERTS) : wbuf;

        // Lane l reads its contiguous 8 experts.
        const float4 xlo = *(const float4*)&buf[8 * lane];
        const float4 xhi = *(const float4*)&buf[8 * lane + 4];

        float sv[8];
        sv[0] = 1.0f / (1.0f + __expf(-xlo.x));
        sv[1] = 1.0f / (1.0f + __expf(-xlo.y));
        sv[2] = 1.0f / (1.0f + __expf(-xlo.z));
        sv[3] = 1.0f / (1.0f + __expf(-xlo.w));
        sv[4] = 1.0f / (1.0f + __expf(-xhi.x));
        sv[5] = 1.0f / (1.0f + __expf(-xhi.y));
        sv[6] = 1.0f / (1.0f + __expf(-xhi.z));
        sv[7] = 1.0f / (1.0f + __expf(-xhi.w));

        // Overwrite the consumed logits with sigmoid scores: exact weight
        // gather later via one ds_load_b32 (LDS ops are in-order per wave).
        *(float4*)&buf[8 * lane]     = make_float4(sv[0], sv[1], sv[2], sv[3]);
        *(float4*)&buf[8 * lane + 4] = make_float4(sv[4], sv[5], sv[6], sv[7]);

        float bv[8];
        bv[0] = sv[0] + blo.x; bv[1] = sv[1] + blo.y;
        bv[2] = sv[2] + blo.z; bv[3] = sv[3] + blo.w;
        bv[4] = sv[4] + bhi.x; bv[5] = sv[5] + bhi.y;
        bv[6] = sv[6] + bhi.z; bv[7] = sv[7] + bhi.w;

        int widx[TOPK];

        #pragma unroll
        for (int k = 0; k < TOPK; ++k) {
            // Lane-local argmax; strict '>' keeps the lowest slot on ties.
            float mv = bv[0]; int ms = 0;
            #pragma unroll
            for (int j = 1; j < 8; ++j) {
                if (bv[j] > mv) { mv = bv[j]; ms = j; }
            }
            // Value-only max butterfly (5 shuffles).
            float g = mv;
            #pragma unroll
            for (int off = 16; off >= 1; off >>= 1) {
                g = fmaxf(g, __shfl_xor(g, off, 32));
            }
            // Winner = lowest lane whose local max equals the global max;
            // lane order == expert-index order, so this matches jax top_k.
            const unsigned long long mk = __ballot(mv == g);
            const int wl = __ffsll(mk) - 1;
            const int ws = __builtin_amdgcn_readlane(ms, wl);  // winner slot (scalar)
            widx[k] = (wl << 3) + ws;
            // Exact single-slot invalidation on the owner lane.
            const bool own = (lane == wl);
            #pragma unroll
            for (int j = 0; j < 8; ++j) {
                if (own && ws == j) bv[j] = NEG_INF;
            }
        }

        // Distribute the 8 (scalar) winner indices: lane group k = lane&7.
        const int s = lane & 7;
        int gidx = widx[0];
        #pragma unroll
        for (int k = 1; k < TOPK; ++k) {
            gidx = (s == k) ? widx[k] : gidx;
        }
        // Single divergent gather of the unbiased sigmoid weights.
        const float w = buf[gidx];
        float sum = w;
        sum += __shfl_xor(sum, 1, 32);
        sum += __shfl_xor(sum, 2, 32);
        sum += __shfl_xor(sum, 4, 32);
        const float inv = 1.0f / (sum + 1e-20f);

        if (lane < TOPK) {
            idx_out[(size_t)cur * TOPK + lane] = gidx;
            w_out[(size_t)cur * TOPK + lane]   = w * inv;
        }
    }
}

extern "C" void kernel_launch(void* const* d_in, const int* in_sizes, int n_in,
                              void* d_out, int out_size, void* d_ws, size_t ws_size,
                              hipStream_t stream) {
    const float* logits = (const float*)d_in[0];   // [T, 256] fp32
    const float* bias   = (const float*)d_in[1];   // [256] fp32
    const int T = in_sizes[0] / EXPERTS;

    // d_out = int32 indices [T,8] followed by float32 weights [T,8].
    int*   idx_out = (int*)d_out;
    float* w_out   = (float*)d_out + (size_t)T * TOPK;

    const int numWaves = (T + TOK_PER_WAVE - 1) / TOK_PER_WAVE;
    const int blocks   = (numWaves + WAVES_PER_BLOCK - 1) / WAVES_PER_BLOCK;
    moe_route_topk8_kernel<<<blocks, 32 * WAVES_PER_BLOCK, 0, stream>>>(
        logits, bias, idx_out, w_out, T);
}